// SystemicGraphFilter_73005854098046
// MI455X (gfx1250) — compile-verified
//
#include <hip/hip_runtime.h>

// SystemicGraphFilter for MI455X (gfx1250, wave32, WGP, WMMA).
// One workgroup (8 waves / 256 threads) per 128x128 batch matrix.
// Whole per-batch working set lives in LDS (~208 KB of the WGP's 320 KB):
//   sC  : clipped+symmetrized correlations, row-major stride 136
//   sS  : raw staging, then soft adjacency, row-major stride 136
//   sFp : filtered, K-pair-interleaved layout (k,col) -> (k>>1)*288 + 2*col + (k&1)
// so a lane's 2-element WMMA B fragment is ONE aligned ds_load_b64 and the
// three graph-propagation GEMMs run on V_WMMA_F32_16X16X4_F32 entirely out of
// LDS, software-pipelined (prefetch next K-step while WMMAs execute).
// Step loop fully unrolled so alpha = {1/2, 1/3, 1/4} are literal constants
// (no v_div_scale/v_div_fixup chain in the hot path).
// HBM traffic is the compulsory 256 MiB (~11 us at 23.3 TB/s).

#define N       128
#define LDSTR   136   // row stride (floats) for sC/sS
#define PSTR    288   // floats per K-row-pair of sFp; 288 mod 64 == 32 -> lane halves in disjoint bank groups
#define NBATCH  2048
#define EPSF    1e-8f
#define NSTEPS  3

typedef __attribute__((ext_vector_type(2))) float v2f;
typedef __attribute__((ext_vector_type(8))) float v8f;

__device__ __forceinline__ int fidx(int row, int col) {
  return (row >> 1) * PSTR + col * 2 + (row & 1);
}

__global__ __launch_bounds__(256, 1)
void sgf_kernel(const float* __restrict__ corr,
                const float* __restrict__ temperature,
                float* __restrict__ out) {
  extern __shared__ float lds[];
  float* sC  = lds;                  // N*LDSTR
  float* sS  = lds + N * LDSTR;      // N*LDSTR
  float* sFp = lds + 2 * N * LDSTR;  // (N/2)*PSTR

  const int tid  = threadIdx.x;
  const int lane = tid & 31;
  const int wave = tid >> 5;       // 0..7 -> output tile-row
  const int lm   = lane & 15;      // lane within half-wave
  const int lh   = lane >> 4;      // lane half (K/M split per WMMA layout)
  const int b    = blockIdx.x;

  const float invT = 1.0f / temperature[0];
  const float* src = corr + (size_t)b * (N * N);

  // ---- stage raw corr[b] into sS (b128 global loads, b128 LDS stores) ----
  for (int e = tid * 4; e < N * N; e += 256 * 4) {
    float4 v = *(const float4*)(src + e);
    int i = e >> 7, j = e & 127;
    *(float4*)&sS[i * LDSTR + j] = v;
  }
  __syncthreads();

  // ---- c = sym(clip(raw, -1+eps, 1-eps)) ----
  for (int e = tid; e < N * N; e += 256) {
    int i = e >> 7, j = e & 127;
    float x = sS[i * LDSTR + j];
    float y = sS[j * LDSTR + i];
    x = fminf(fmaxf(x, -1.0f + EPSF), 1.0f - EPSF);
    y = fminf(fmaxf(y, -1.0f + EPSF), 1.0f - EPSF);
    sC[i * LDSTR + j] = 0.5f * (x + y);
  }
  __syncthreads();

  // ---- affinities, top-3 threshold, masked softmax (one row per lane) ----
  if (tid < N) {
    const int i = tid;
    float t0 = -1e30f, t1 = -1e30f, t2 = -1e30f;
    for (int j = 0; j < N; ++j) {
      float cv  = sC[i * LDSTR + j];
      float d   = __builtin_sqrtf(2.0f * (1.0f - cv) + EPSF);
      float aff = __expf(-d * invT);
      if (j == i) aff = 0.0f;          // affinities * (1 - eye)
      sS[i * LDSTR + j] = aff;
      if (aff > t0)      { t2 = t1; t1 = t0; t0 = aff; }
      else if (aff > t1) { t2 = t1; t1 = aff; }
      else if (aff > t2) { t2 = aff; }
    }
    // softmax over {aff >= t2}; masked entries are exactly 0 (exp(-1e9) == 0)
    float sum = 0.0f;
    for (int j = 0; j < N; ++j) {
      float a = sS[i * LDSTR + j];
      if (a >= t2) sum += __expf(a - t0);
    }
    float rsum = 1.0f / sum;
    for (int j = 0; j < N; ++j) {
      float a = sS[i * LDSTR + j];
      sS[i * LDSTR + j] = (a >= t2) ? __expf(a - t0) * rsum : 0.0f;
    }
  }
  __syncthreads();

  // ---- symmetrize soft_adj in place ----
  for (int e = tid; e < N * N; e += 256) {
    int i = e >> 7, j = e & 127;
    if (i < j) {
      float a = sS[i * LDSTR + j];
      float c2 = sS[j * LDSTR + i];
      float m = 0.5f * (a + c2);
      sS[i * LDSTR + j] = m;
      sS[j * LDSTR + i] = m;
    }
  }

  // ---- filtered = c (into pair-interleaved layout) ----
  for (int e = tid; e < N * N; e += 256) {
    int i = e >> 7, j = e & 127;
    sFp[fidx(i, j)] = sC[i * LDSTR + j];
  }
  __syncthreads();

  // ---- 3 rounds of messages = soft_adj @ filtered via V_WMMA_F32_16X16X4 ----
  // Wave w owns output rows [16w, 16w+16) as 8 accumulator tiles (v8f each).
  const int tm       = wave;
  const int arowbase = (tm * 16 + lm) * LDSTR + 2 * lh;  // A: M = lane%16, K pair = 2*lh
  const int bcolbase = lh * PSTR + lm * 2;               // B: pairIdx = k/2 + lh, col = tn*16+lm

  const float kAlpha[NSTEPS] = { 0.5f, 1.0f / 3.0f, 0.25f };  // 1/(step+2)

#pragma unroll
  for (int step = 0; step < NSTEPS; ++step) {
    v8f acc[8] = {};

    // prologue: fragments for k = 0
    v2f av = *(const v2f*)&sS[arowbase];
    v2f bcur[8];
#pragma unroll
    for (int tn = 0; tn < 8; ++tn)
      bcur[tn] = *(const v2f*)&sFp[bcolbase + tn * 32];

    // steady state: prefetch k+4 while WMMAs for k execute
    for (int k = 0; k < N - 4; k += 4) {
      v2f an = *(const v2f*)&sS[arowbase + k + 4];
      const int bb = ((k + 4) >> 1) * PSTR + bcolbase;
      v2f bn[8];
#pragma unroll
      for (int tn = 0; tn < 8; ++tn)
        bn[tn] = *(const v2f*)&sFp[bb + tn * 32];
#pragma unroll
      for (int tn = 0; tn < 8; ++tn)
        acc[tn] = __builtin_amdgcn_wmma_f32_16x16x4_f32(
            false, av, false, bcur[tn], (short)0, acc[tn], false, false);
      av = an;
#pragma unroll
      for (int tn = 0; tn < 8; ++tn) bcur[tn] = bn[tn];
    }
    // epilogue: last K-step
#pragma unroll
    for (int tn = 0; tn < 8; ++tn)
      acc[tn] = __builtin_amdgcn_wmma_f32_16x16x4_f32(
          false, av, false, bcur[tn], (short)0, acc[tn], false, false);

    __syncthreads();   // all waves done reading sFp -> safe to update in place

    const float alpha = kAlpha[step];   // compile-time constant after unroll
#pragma unroll
    for (int tn = 0; tn < 8; ++tn) {
#pragma unroll
      for (int v = 0; v < 8; ++v) {
        // C/D layout: M = v + 8*lanehalf, Nloc = lane%16
        int row  = tm * 16 + v + 8 * lh;
        int col  = tn * 16 + lm;
        int ridx = row * LDSTR + col;
        int fi   = fidx(row, col);
        float msg = acc[tn][v];
        float f   = sFp[fi];
        float s   = sS[ridx];
        float cc  = sC[ridx];
        float fn  = __builtin_fmaf(alpha, msg - f, f);   // (1-a)*f + a*msg
        sFp[fi]   = __builtin_fmaf(s, cc - fn, fn);      // s*c + (1-s)*fn
      }
    }
    __syncthreads();
  }

  // ---- out = sym(filtered) off-diagonal, 1 on diagonal ----
  float* dst = out + (size_t)b * (N * N);
  for (int e = tid; e < N * N; e += 256) {
    int i = e >> 7, j = e & 127;
    float val = (i == j) ? 1.0f
                         : 0.5f * (sFp[fidx(i, j)] + sFp[fidx(j, i)]);
    dst[e] = val;
  }
}

extern "C" void kernel_launch(void* const* d_in, const int* in_sizes, int n_in,
                              void* d_out, int out_size, void* d_ws, size_t ws_size,
                              hipStream_t stream) {
  (void)in_sizes; (void)n_in; (void)out_size; (void)d_ws; (void)ws_size;
  const float* corr = (const float*)d_in[0];
  const float* temp = (const float*)d_in[1];
  float* out = (float*)d_out;

  const size_t shmem =
      ((size_t)2 * N * LDSTR + (size_t)(N / 2) * PSTR) * sizeof(float);  // 212,992 B < 320 KB
  static bool attr_set = false;   // idempotent, deterministic; not a stream op
  if (!attr_set) {
    hipFuncSetAttribute((const void*)sgf_kernel,
                        hipFuncAttributeMaxDynamicSharedMemorySize, (int)shmem);
    attr_set = true;
  }
  sgf_kernel<<<dim3(NBATCH), dim3(256), shmem, stream>>>(corr, temp, out);
}